// VariationalGCNEncoder_44341242364729
// MI455X (gfx1250) — compile-verified
//
#include <hip/hip_runtime.h>

#define CH 128        // IN_C == HID == 128
#define OC 64         // OUT_C

typedef __attribute__((ext_vector_type(2))) float v2f;
typedef __attribute__((ext_vector_type(4))) float v4f;
typedef __attribute__((ext_vector_type(8))) float v8f;

// Hardware FP32 scatter-add (GLOBAL_ATOMIC_ADD_F32, non-returning, GV mode).
__device__ __forceinline__ void atomAddF(float* p, float v) {
  asm volatile("global_atomic_add_f32 %0, %1, off" :: "v"(p), "v"(v) : "memory");
}

// ---------------- degree / normalization ----------------
__global__ void k_init_cnt(int* cnt, int n) {
  int i = blockIdx.x * blockDim.x + threadIdx.x;
  if (i < n) cnt[i] = 0;
}

__global__ void k_count(const int* __restrict__ dst, int* cnt, int e) {
  int i = blockIdx.x * blockDim.x + threadIdx.x;
  if (i < e) atomicAdd(&cnt[dst[i]], 1);
}

// dinv[i] = rsqrt(deg_with_self_loop) ; cnt and dinv alias the same buffer
__global__ void k_dinv(const int* __restrict__ cnt, float* dinv, int n) {
  int i = blockIdx.x * blockDim.x + threadIdx.x;
  if (i < n) {
    float deg = (float)(cnt[i] + 1);   // +1 self loop; always > 0
    dinv[i] = rsqrtf(deg);
  }
}

// ---------------- aggregation ----------------
// out[i][c] = feat[i][c] * dinv[i]^2   (self-loop term; also zero-inits acc)
__global__ void k_selfinit(const float* __restrict__ feat,
                           const float* __restrict__ dinv,
                           float* __restrict__ out, int n) {
  long i = (long)blockIdx.x * blockDim.x + threadIdx.x;
  long total = (long)n * CH;
  if (i >= total) return;
  int row = (int)(i >> 7);
  float di = dinv[row];
  out[i] = feat[i] * di * di;
}

// acc[dst][*] += feat[src][*] * dinv[src]*dinv[dst]; one wave per edge,
// 4 channels per lane. Edge id is wave-uniform -> scalarize index loads.
__global__ void k_agg(const float* __restrict__ feat,
                      const int* __restrict__ src,
                      const int* __restrict__ dst,
                      const float* __restrict__ dinv,
                      float* __restrict__ acc, int e) {
  long gid = (long)blockIdx.x * blockDim.x + threadIdx.x;
  int ed = __builtin_amdgcn_readfirstlane((int)(gid >> 5));
  int lane = threadIdx.x & 31;
  if (ed >= e) return;
  int s = __builtin_amdgcn_readfirstlane(src[ed]);
  int d = __builtin_amdgcn_readfirstlane(dst[ed]);
  float w = dinv[s] * dinv[d];
  v4f v = *(const v4f*)(feat + (long)s * CH + lane * 4);
  float* ap = acc + (long)d * CH + lane * 4;
  atomAddF(ap + 0, v.x * w);
  atomAddF(ap + 1, v.y * w);
  atomAddF(ap + 2, v.z * w);
  atomAddF(ap + 3, v.w * w);
}

// ---------------- WMMA GEMMs (fp32, 16x16x4) ----------------
// Stage one 16x128 A-tile into LDS (padded stride 132 -> no bank conflicts).
__device__ __forceinline__ void stage_tile(const float* __restrict__ A,
                                           float* As, int rowbase, int n, int tid) {
  int r  = tid >> 4;            // row within tile (8 floats per thread)
  int k0 = (tid * 8) & 127;
  int gr = rowbase + r;
  v4f p0 = {0.f, 0.f, 0.f, 0.f}, p1 = p0;
  if (gr < n) {
    const float* g = A + (long)gr * CH + k0;
    p0 = *(const v4f*)(g);
    p1 = *(const v4f*)(g + 4);
  }
  *(v4f*)&As[r * 132 + k0]     = p0;
  *(v4f*)&As[r * 132 + k0 + 4] = p1;
}

// H = relu(A @ W + bias), A:[n,128], W:[128,128]. One block = 16-row tile,
// 8 waves cover the 8 column tiles.
__global__ __launch_bounds__(256) void k_gemm_relu(const float* __restrict__ A,
                                                   const float* __restrict__ W,
                                                   const float* __restrict__ bias,
                                                   float* __restrict__ H, int n) {
  __shared__ float As[16 * 132];
  int tid = threadIdx.x;
  int rowbase = blockIdx.x * 16;
  stage_tile(A, As, rowbase, n, tid);
  __syncthreads();

  int wv   = tid >> 5;
  int L    = tid & 31;
  int half = L >> 4;        // 0: K pair {0,1}, 1: K pair {2,3} of each 4-chunk
  int l15  = L & 15;
  int col  = wv * 16 + l15;

  v8f acc = {0.f, 0.f, 0.f, 0.f, 0.f, 0.f, 0.f, 0.f};
  const float* arow = &As[l15 * 132 + half * 2];
  const float* wcol = W + col;                    // row stride 128
#pragma unroll
  for (int kk = 0; kk < 32; ++kk) {
    int k = kk * 4;
    v2f af; af.x = arow[k];                 af.y = arow[k + 1];
    v2f bf; bf.x = wcol[(k + half * 2) * CH]; bf.y = wcol[(k + half * 2 + 1) * CH];
    acc = __builtin_amdgcn_wmma_f32_16x16x4_f32(false, af, false, bf,
                                                (short)0, acc, false, false);
  }
  float bc = bias[col];
#pragma unroll
  for (int i = 0; i < 8; ++i) {
    int row = rowbase + i + half * 8;
    if (row < n) {
      float v = acc[i] + bc;
      H[(long)row * CH + col] = v > 0.f ? v : 0.f;
    }
  }
}

// [mu | logstd] in one kernel: waves 0-3 -> A@Wm + bm, waves 4-7 -> A@Wl + bl.
__global__ __launch_bounds__(256) void k_gemm_out(const float* __restrict__ A,
                                                  const float* __restrict__ Wm,
                                                  const float* __restrict__ bm,
                                                  const float* __restrict__ Wl,
                                                  const float* __restrict__ bl,
                                                  float* __restrict__ out, int n) {
  __shared__ float As[16 * 132];
  int tid = threadIdx.x;
  int rowbase = blockIdx.x * 16;
  stage_tile(A, As, rowbase, n, tid);
  __syncthreads();

  int wv   = tid >> 5;
  int isls = wv >> 2;       // 0 = mu, 1 = logstd
  int ct   = wv & 3;
  int L    = tid & 31;
  int half = L >> 4;
  int l15  = L & 15;
  int col  = ct * 16 + l15;

  const float* W = isls ? Wl : Wm;
  const float* B = isls ? bl : bm;
  float* O = out + (long)isls * (long)n * OC;

  v8f acc = {0.f, 0.f, 0.f, 0.f, 0.f, 0.f, 0.f, 0.f};
  const float* arow = &As[l15 * 132 + half * 2];
  const float* wcol = W + col;                    // row stride 64
#pragma unroll
  for (int kk = 0; kk < 32; ++kk) {
    int k = kk * 4;
    v2f af; af.x = arow[k];                 af.y = arow[k + 1];
    v2f bf; bf.x = wcol[(k + half * 2) * OC]; bf.y = wcol[(k + half * 2 + 1) * OC];
    acc = __builtin_amdgcn_wmma_f32_16x16x4_f32(false, af, false, bf,
                                                (short)0, acc, false, false);
  }
  float bc = B[col];
#pragma unroll
  for (int i = 0; i < 8; ++i) {
    int row = rowbase + i + half * 8;
    if (row < n) O[(long)row * OC + col] = acc[i] + bc;
  }
}

// ---------------- launch ----------------
extern "C" void kernel_launch(void* const* d_in, const int* in_sizes, int n_in,
                              void* d_out, int out_size, void* d_ws, size_t ws_size,
                              hipStream_t stream) {
  const float* x   = (const float*)d_in[0];
  const int*   ei  = (const int*)d_in[1];
  const float* w1  = (const float*)d_in[2];
  const float* b1  = (const float*)d_in[3];
  const float* wmu = (const float*)d_in[4];
  const float* bmu = (const float*)d_in[5];
  const float* wls = (const float*)d_in[6];
  const float* bls = (const float*)d_in[7];

  int n = in_sizes[0] / CH;
  int e = in_sizes[1] / 2;
  const int* src = ei;          // edge_index[0]
  const int* dst = ei + e;      // edge_index[1]

  float* ws   = (float*)d_ws;
  float* dinv = ws;                              // n floats (int counts first)
  float* a    = ws + (1 << 17);                  // n*128 aggregation buffer
  float* h    = a + (long)n * CH;                // n*128 hidden

  const int TB = 256;
  long nc = (long)n * CH;
  long et = (long)e * 32;

  // normalization
  k_init_cnt<<<(n + TB - 1) / TB, TB, 0, stream>>>((int*)dinv, n);
  k_count   <<<(e + TB - 1) / TB, TB, 0, stream>>>(dst, (int*)dinv, e);
  k_dinv    <<<(n + TB - 1) / TB, TB, 0, stream>>>((const int*)dinv, dinv, n);

  // layer 1: a = Agg(x); h = relu(a @ w1 + b1)
  k_selfinit<<<(int)((nc + TB - 1) / TB), TB, 0, stream>>>(x, dinv, a, n);
  k_agg     <<<(int)((et + TB - 1) / TB), TB, 0, stream>>>(x, src, dst, dinv, a, e);
  k_gemm_relu<<<(n + 15) / 16, TB, 0, stream>>>(a, w1, b1, h, n);

  // layer 2: a = Agg(h); [mu | logstd] = a @ [w_mu | w_ls] + bias -> d_out
  k_selfinit<<<(int)((nc + TB - 1) / TB), TB, 0, stream>>>(h, dinv, a, n);
  k_agg     <<<(int)((et + TB - 1) / TB), TB, 0, stream>>>(h, src, dst, dinv, a, e);
  k_gemm_out<<<(n + 15) / 16, TB, 0, stream>>>(a, wmu, bmu, wls, bls, (float*)d_out, n);
}